// Generator4DLUT_identity_37306085933294
// MI455X (gfx1250) — compile-verified
//
#include <hip/hip_runtime.h>
#include <hip/hip_bf16.h>
#include <cstdint>

#define DIMV   33
#define D2     (DIMV * DIMV)            // 1089
#define D3     (DIMV * DIMV * DIMV)     // 35937
#define LUTN   (2 * D3)                 // 71874 entries per channel
#define PLANE  (512 * 512)              // 262144
#define NPIX   (8 * PLANE)              // 2097152 pixels
#define BLK    256

// ---------------------------------------------------------------------------
// One-time (per launch) AoS packing of the LUT: packed[i] = {lutC0[i], lutC1[i],
// lutC2[i], 0}. Turns 3 scattered B32 gathers per corner into one B128 gather.
// ---------------------------------------------------------------------------
__global__ __launch_bounds__(BLK)
void lut_pack_kernel(const float* __restrict__ lut, float4* __restrict__ packed) {
    int i = blockIdx.x * BLK + threadIdx.x;
    if (i >= LUTN) return;
    float4 v;
    v.x = lut[i];
    v.y = lut[i + LUTN];
    v.z = lut[i + 2 * LUTN];
    v.w = 0.0f;
    packed[i] = v;
}

// ---------------------------------------------------------------------------
// Main quadrilinear-4D LUT kernel. One thread = one pixel.
// Coordinates are staged plane-by-plane into LDS with CDNA5 async loads
// (GLOBAL_LOAD_ASYNC_TO_LDS_B32, tracked by ASYNCcnt).
// ---------------------------------------------------------------------------
template <bool PACKED>
__global__ __launch_bounds__(BLK)
void lut4d_kernel(const float* __restrict__ x,
                  const float* __restrict__ lut,     // SoA fallback
                  const float4* __restrict__ lutp,   // AoS packed
                  float* __restrict__ out) {
    __shared__ float tileX[4 * BLK];

    const int tid = threadIdx.x;
    const long long pix = (long long)blockIdx.x * BLK + tid;
    const int b  = (int)(pix >> 18);           // / PLANE
    const int sp = (int)(pix & (PLANE - 1));   // % PLANE

    // ---- async-stage the 4 coordinate planes for this tile into LDS ----
    {
        const unsigned ldsBase = (unsigned)(uintptr_t)(&tileX[0]);
        const float* gbase = x + ((size_t)b * 4) * PLANE + sp;
#pragma unroll
        for (int ch = 0; ch < 4; ++ch) {
            unsigned la = ldsBase + (unsigned)((ch * BLK + tid) * 4);
            unsigned long long ga =
                (unsigned long long)(uintptr_t)(gbase + (size_t)ch * PLANE);
            asm volatile("global_load_async_to_lds_b32 %0, %1, off"
                         :: "v"(la), "v"(ga) : "memory");
        }
        asm volatile("s_wait_asynccnt 0x0" ::: "memory");
    }

    const float xc0 = tileX[0 * BLK + tid];
    const float xc1 = tileX[1 * BLK + tid];
    const float xc2 = tileX[2 * BLK + tid];
    const float xc3 = tileX[3 * BLK + tid];

    const float binsize = (float)(1.000001 / 32.0);
    float s0 = xc0 / binsize, s1 = xc1 / binsize, s2 = xc2 / binsize, s3 = xc3 / binsize;

    float fl0 = floorf(s0), fl1 = floorf(s1), fl2 = floorf(s2), fl3 = floorf(s3);
    int   i0 = (int)fl0,    i1 = (int)fl1,    i2 = (int)fl2,    i3 = (int)fl3;
    float f0 = s0 - fl0,    f1 = s1 - fl1,    f2 = s2 - fl2,    f3 = s3 - fl3;

    // two-level weight factorization: w[p] = hi[p>>2] * lo[p&3]
    float a0 = 1.0f - f0, a1 = 1.0f - f1, a2 = 1.0f - f2, a3 = 1.0f - f3;
    float hi[4] = { a0 * a1, a0 * f1, f0 * a1, f0 * f1 };   // dims 0,1
    float lo[4] = { a2 * a3, a2 * f3, f2 * a3, f2 * f3 };   // dims 2,3
    const int ohi[4] = { 0, D2, D3, D3 + D2 };
    const int olo[4] = { 0, 1, DIMV, DIMV + 1 };

    const int base = i0 * D3 + i1 * D2 + i2 * DIMV + i3;

    float ax = 0.0f, ay = 0.0f, az = 0.0f;
#pragma unroll
    for (int jh = 0; jh < 4; ++jh) {
        const int bh = base + ohi[jh];
#pragma unroll
        for (int jl = 0; jl < 4; ++jl) {
            const float wt = hi[jh] * lo[jl];
            if (PACKED) {
                float4 v = lutp[bh + olo[jl]];            // one B128 gather
                ax = fmaf(wt, v.x, ax);
                ay = fmaf(wt, v.y, ay);
                az = fmaf(wt, v.z, az);
            } else {
                const float* p = lut + (bh + olo[jl]);    // 3 B32 gathers
                ax = fmaf(wt, p[0],        ax);
                ay = fmaf(wt, p[LUTN],     ay);
                az = fmaf(wt, p[2 * LUTN], az);
            }
        }
    }

    float* ob = out + ((size_t)b * 3) * PLANE + sp;
    __builtin_nontemporal_store(ax, ob);
    __builtin_nontemporal_store(ay, ob + PLANE);
    __builtin_nontemporal_store(az, ob + 2 * PLANE);
}

extern "C" void kernel_launch(void* const* d_in, const int* in_sizes, int n_in,
                              void* d_out, int out_size, void* d_ws, size_t ws_size,
                              hipStream_t stream) {
    (void)in_sizes; (void)n_in; (void)out_size;
    const float* lut = (const float*)d_in[0];   // (3, 2, 33, 33, 33) f32
    const float* x   = (const float*)d_in[1];   // (8, 4, 512, 512)   f32
    float* out = (float*)d_out;                 // (8, 3, 512, 512)   f32

    const size_t packBytes = (size_t)LUTN * sizeof(float4);
    const int nPixBlocks = NPIX / BLK;          // 8192

    if (d_ws && ws_size >= packBytes) {
        const int nPackBlocks = (LUTN + BLK - 1) / BLK;
        lut_pack_kernel<<<nPackBlocks, BLK, 0, stream>>>(lut, (float4*)d_ws);
        lut4d_kernel<true><<<nPixBlocks, BLK, 0, stream>>>(
            x, lut, (const float4*)d_ws, out);
    } else {
        lut4d_kernel<false><<<nPixBlocks, BLK, 0, stream>>>(
            x, lut, nullptr, out);
    }
}